// VoxelizationOp_19112604467258
// MI455X (gfx1250) — compile-verified
//
#include <hip/hip_runtime.h>
#include <hip/hip_bf16.h>

#define GX 432
#define GY 496
#define GZ 1
#define NVOX (GX * GY * GZ)   /* 214272 = 837 * 256 exactly */
#define BLK 256
#define NBLK (NVOX / BLK)     /* 837 */

#define TILE_PTS 1024
#define TILE_FLOATS (TILE_PTS * 5)        /* 5120 floats = 20 KB */
#define TILE_CHUNKS (TILE_PTS * 20 / 16)  /* 1280 b128 chunks */
#define CHUNKS_PER_THREAD (TILE_CHUNKS / BLK) /* 5 -> 5 async issues per wave per tile */

// ---------------------------------------------------------------------------
// Zero the accumulator region of the workspace (sums as f32=0 bits, counts=0)
// ---------------------------------------------------------------------------
__global__ void vox_zero_ws(int* __restrict__ w, int n) {
  int i = blockIdx.x * blockDim.x + threadIdx.x;
  if (i < n) w[i] = 0;
}

// ---------------------------------------------------------------------------
// Default-fill the output: feats=0, coords=-1, cnt=0 (unoccupied slots keep
// these; occupied slots are overwritten by the compaction pass).
// ---------------------------------------------------------------------------
__global__ void vox_fill_out(float* __restrict__ out, int coordBase, int cntBase, int total) {
  int i = blockIdx.x * blockDim.x + threadIdx.x;
  if (i < total) out[i] = (i >= coordBase && i < cntBase) ? -1.0f : 0.0f;
}

// ---------------------------------------------------------------------------
// Scatter pass: double-buffered async staging of the point stream through LDS
// (global_load_async_to_lds_b128, ASYNCcnt-tracked) followed by atomic
// accumulation into the L2-resident per-voxel sums/counts.
//
// Each wave issues exactly CHUNKS_PER_THREAD (5) uniform async ops per tile,
// so "s_wait_asynccnt 5" retires exactly the previous tile while the next
// tile's DMA is in flight. Tail chunks are clamped to the last valid 16B
// chunk (junk lands in unused LDS); points whose 20B row is not fully
// covered by valid chunks (only the final point) use direct global loads.
// ---------------------------------------------------------------------------
__global__ void vox_scatter(const float* __restrict__ pts,
                            float* __restrict__ sums,
                            int* __restrict__ counts,
                            long n) {
  __shared__ float tile[2][TILE_FLOATS];   /* 40 KB of the 320 KB WGP LDS */

  const long nBytes = n * 20l;
  const long maxChunkByte = nBytes - 16;   /* clamp target; n >= 1 point */
  const long nTiles = (n + TILE_PTS - 1) / TILE_PTS;
  const uintptr_t gbase = (uintptr_t)pts;

  long t = blockIdx.x;
  int cur = 0;

  // Prologue: kick off DMA for this block's first tile.
  if (t < nTiles) {
    const long tb = t * (long)(TILE_PTS * 20);
#pragma unroll
    for (int k = 0; k < CHUNKS_PER_THREAD; ++k) {
      const int c = threadIdx.x + k * BLK;
      long gb = tb + 16l * c;
      if (gb > maxChunkByte) gb = maxChunkByte;      /* uniform issue count */
      const unsigned long long ga = (unsigned long long)(gbase + (uintptr_t)gb);
      const unsigned la = (unsigned)(uintptr_t)&tile[0][c * 4];
      asm volatile("global_load_async_to_lds_b128 %0, %1, off"
                   :: "v"(la), "v"(ga) : "memory");
    }
  }

  for (; t < nTiles; t += gridDim.x) {
    const long tn = t + (long)gridDim.x;
    if (tn < nTiles) {
      // Issue next tile's DMA into the other buffer, then wait for current.
      const long tb = tn * (long)(TILE_PTS * 20);
#pragma unroll
      for (int k = 0; k < CHUNKS_PER_THREAD; ++k) {
        const int c = threadIdx.x + k * BLK;
        long gb = tb + 16l * c;
        if (gb > maxChunkByte) gb = maxChunkByte;
        const unsigned long long ga = (unsigned long long)(gbase + (uintptr_t)gb);
        const unsigned la = (unsigned)(uintptr_t)&tile[cur ^ 1][c * 4];
        asm volatile("global_load_async_to_lds_b128 %0, %1, off"
                     :: "v"(la), "v"(ga) : "memory");
      }
      asm volatile("s_wait_asynccnt 0x5" ::: "memory"); /* next tile may stay in flight */
    } else {
      asm volatile("s_wait_asynccnt 0x0" ::: "memory");
    }
    __syncthreads();

    // Process the current tile: 4 points per thread from LDS (stride-5 dword
    // reads are conflict-free across 64 banks: gcd(5,64)=1).
#pragma unroll
    for (int k = 0; k < TILE_PTS / BLK; ++k) {
      const int lp = threadIdx.x + k * BLK;
      const long p = t * (long)TILE_PTS + lp;
      if (p < n) {
        float x, y, z, a, b;
        const long bp = p * 20l;
        if (bp + 35 <= nBytes) {                 /* fully covered by valid chunks */
          const float* r = &tile[cur][lp * 5];
          x = r[0]; y = r[1]; z = r[2]; a = r[3]; b = r[4];
        } else {                                  /* only the very last point */
          x = pts[5 * p + 0]; y = pts[5 * p + 1]; z = pts[5 * p + 2];
          a = pts[5 * p + 3]; b = pts[5 * p + 4];
        }

        // Match reference: floor((p - pc_min) / voxel_size) in f32, int cast.
        const int cx = (int)floorf((x - 0.0f)   / 0.16f);
        const int cy = (int)floorf((y + 39.68f) / 0.16f);
        const int cz = (int)floorf((z + 3.0f)   / 4.0f);
        const bool valid = (cx >= 0) & (cy >= 0) & (cz >= 0) &
                           (cx < GX) & (cy < GY) & (cz < GZ);
        if (valid) {
          const int lin = (cz * GY + cy) * GX + cx;
          float* s = sums + 5l * lin;
          atomicAdd(s + 0, x);
          atomicAdd(s + 1, y);
          atomicAdd(s + 2, z);
          atomicAdd(s + 3, a);
          atomicAdd(s + 4, b);
          atomicAdd(counts + lin, 1);
        }
      }
    }
    __syncthreads();   /* all waves done reading before buffer reuse */
    cur ^= 1;
  }
}

// ---------------------------------------------------------------------------
// Per-block count of non-empty voxels (wave32 ballot reduction; 256 voxels
// per block, NVOX is an exact multiple of 256 so no tail handling).
// ---------------------------------------------------------------------------
__global__ void vox_block_count(const int* __restrict__ counts,
                                int* __restrict__ blockCounts) {
  __shared__ int wsum[BLK / 32];
  const int v = blockIdx.x * BLK + threadIdx.x;
  const int flag = counts[v] > 0 ? 1 : 0;
  const unsigned bal = (unsigned)__ballot(flag);   // wave32: low 32 bits
  const int lane = threadIdx.x & 31;
  const int wid  = threadIdx.x >> 5;
  if (lane == 0) wsum[wid] = __popc(bal);
  __syncthreads();
  if (threadIdx.x == 0) {
    int t = 0;
#pragma unroll
    for (int i = 0; i < BLK / 32; ++i) t += wsum[i];
    blockCounts[blockIdx.x] = t;
  }
}

// ---------------------------------------------------------------------------
// Single-workgroup exclusive scan over the 837 block counts (Hillis-Steele
// in LDS, 1024 threads >= NBLK).
// ---------------------------------------------------------------------------
__global__ void vox_scan_blocks(const int* __restrict__ blockCounts,
                                int* __restrict__ blockOffsets,
                                int nblk) {
  __shared__ int sm[1024];
  const int t = threadIdx.x;
  const int v = (t < nblk) ? blockCounts[t] : 0;
  sm[t] = v;
  __syncthreads();
  for (int off = 1; off < 1024; off <<= 1) {
    const int add = (t >= off) ? sm[t - off] : 0;
    __syncthreads();
    sm[t] += add;
    __syncthreads();
  }
  if (t < nblk) blockOffsets[t] = sm[t] - v;  // exclusive prefix
}

// ---------------------------------------------------------------------------
// Stable compaction: wave32 ballot prefix within block + LDS scan of the
// 8 per-wave totals + scanned global block offset -> stable output slot.
// Writes feats = sums/cnt, coords = (z,y,x), cnt for occupied slots < maxV.
// ---------------------------------------------------------------------------
__global__ void vox_compact(const float* __restrict__ sums,
                            const int* __restrict__ counts,
                            const int* __restrict__ blockOffsets,
                            float* __restrict__ out,
                            int maxV) {
  __shared__ int wbase[BLK / 32];
  const int v = blockIdx.x * BLK + threadIdx.x;
  const int c = counts[v];
  const int flag = c > 0 ? 1 : 0;
  const unsigned bal = (unsigned)__ballot(flag);
  const int lane = threadIdx.x & 31;
  const int wid  = threadIdx.x >> 5;
  if (lane == 0) wbase[wid] = __popc(bal);
  __syncthreads();
  if (threadIdx.x == 0) {
    int acc = 0;
#pragma unroll
    for (int i = 0; i < BLK / 32; ++i) { const int x = wbase[i]; wbase[i] = acc; acc += x; }
  }
  __syncthreads();
  if (flag) {
    const int local = wbase[wid] + __popc(bal & ((1u << lane) - 1u));
    const int pos = blockOffsets[blockIdx.x] + local;
    if (pos < maxV) {
      const float fc = (float)c;
      const float* s = sums + 5l * v;
      float* f = out + 5l * pos;
      f[0] = s[0] / fc;
      f[1] = s[1] / fc;
      f[2] = s[2] / fc;
      f[3] = s[3] / fc;
      f[4] = s[4] / fc;

      const int zi  = v / (GX * GY);
      const int rem = v % (GX * GY);
      const int yi  = rem / GX;
      const int xi  = rem % GX;
      float* co = out + 5l * maxV + 3l * pos;
      co[0] = (float)zi;
      co[1] = (float)yi;
      co[2] = (float)xi;

      out[8l * maxV + pos] = fc;
    }
  }
}

extern "C" void kernel_launch(void* const* d_in, const int* in_sizes, int n_in,
                              void* d_out, int out_size, void* d_ws, size_t ws_size,
                              hipStream_t stream) {
  (void)n_in; (void)ws_size;

  const float* pts = (const float*)d_in[0];
  const long n = (long)in_sizes[0] / 5;          // 4,000,000 points
  const int maxV = out_size / 9;                  // 40,000 (5 feat + 3 coord + 1 cnt)
  float* out = (float*)d_out;

  // Workspace layout: sums[NV*5] f32 | counts[NV] i32 | blkCnt[NBLK] | blkOff[NBLK]
  float* sums         = (float*)d_ws;
  int*   counts       = (int*)(sums + 5l * NVOX);
  int*   blockCounts  = counts + NVOX;
  int*   blockOffsets = blockCounts + NBLK;

  // 1) Zero accumulator (sums + counts = 6*NV dwords).
  const int zeroN = NVOX * 6;
  vox_zero_ws<<<(zeroN + 255) / 256, 256, 0, stream>>>((int*)d_ws, zeroN);

  // 2) Default-fill output (feats=0, coords=-1, cnt=0).
  vox_fill_out<<<(out_size + 255) / 256, 256, 0, stream>>>(out, 5 * maxV, 8 * maxV, out_size);

  // 3) Scatter-accumulate points (async-LDS staged, double-buffered).
  vox_scatter<<<512, BLK, 0, stream>>>(pts, sums, counts, n);

  // 4) Per-block non-empty counts.
  vox_block_count<<<NBLK, BLK, 0, stream>>>(counts, blockCounts);

  // 5) Exclusive scan of block counts (single workgroup).
  vox_scan_blocks<<<1, 1024, 0, stream>>>(blockCounts, blockOffsets, NBLK);

  // 6) Stable compaction into the first maxV slots.
  vox_compact<<<NBLK, BLK, 0, stream>>>(sums, counts, blockOffsets, out, maxV);
}